// FPModule_7842610283205
// MI455X (gfx1250) — compile-verified
//
#include <hip/hip_runtime.h>
#include <hip/hip_bf16.h>

typedef float v2f __attribute__((ext_vector_type(2)));
typedef float v4f __attribute__((ext_vector_type(4)));
typedef float v8f __attribute__((ext_vector_type(8)));

#define EPSV 1e-7f

// ---------------------------------------------------------------------------
// Kernel 1: per query point, 3-NN over the 1024 source points (xyz1 staged in
// LDS), inverse-distance-weighted interpolation of 256-ch features, concat
// with the query's own 128-ch features into X (row-major 32768 x 384), and
// pass xyz2 through to the output tail.
// grid = (4096/256, B), block = 256 (8 waves)
// ---------------------------------------------------------------------------
__global__ void __launch_bounds__(256) knn_interp_kernel(
    const float* __restrict__ pc1,   // (B,1024,259)
    const float* __restrict__ pc2,   // (B,4096,131)
    float* __restrict__ X,           // (B*4096, 384)
    float* __restrict__ out_xyz)     // (B*4096, 3)
{
    __shared__ float sx[1024], sy[1024], sz[1024];
    const int b   = blockIdx.y;
    const int tid = threadIdx.x;
    const int q   = blockIdx.x * 256 + tid;

    const float* p1 = pc1 + (size_t)b * 1024 * 259;
    for (int i = tid; i < 1024; i += 256) {
        const float* r = p1 + (size_t)i * 259;
        sx[i] = r[0]; sy[i] = r[1]; sz[i] = r[2];
    }
    __syncthreads();

    const float* qrow = pc2 + ((size_t)b * 4096 + q) * 131;
    const float qx = qrow[0], qy = qrow[1], qz = qrow[2];

    // running top-3 smallest squared distances (ties keep earliest index,
    // matching top_k ordering)
    float b0 = 1e30f, b1 = 1e30f, b2 = 1e30f;
    int   i0 = 0,     i1 = 0,     i2 = 0;
    for (int i = 0; i < 1024; ++i) {
        float dx = sx[i] - qx, dy = sy[i] - qy, dz = sz[i] - qz;
        float d  = dx * dx + dy * dy + dz * dz;
        if (d < b2) {
            if (d < b1) {
                b2 = b1; i2 = i1;
                if (d < b0) { b1 = b0; i1 = i0; b0 = d; i0 = i; }
                else        { b1 = d;  i1 = i; }
            } else { b2 = d; i2 = i; }
        }
    }
    const float r0 = 1.0f / fmaxf(b0, EPSV);
    const float r1 = 1.0f / fmaxf(b1, EPSV);
    const float r2 = 1.0f / fmaxf(b2, EPSV);
    const float inv_norm = 1.0f / (r0 + r1 + r2);
    const float w0 = r0 * inv_norm, w1 = r1 * inv_norm, w2 = r2 * inv_norm;

    const float* f0 = p1 + (size_t)i0 * 259 + 3;
    const float* f1 = p1 + (size_t)i1 * 259 + 3;
    const float* f2 = p1 + (size_t)i2 * 259 + 3;
    float* xrow = X + ((size_t)b * 4096 + q) * 384;
#pragma unroll 8
    for (int c = 0; c < 256; ++c)
        xrow[c] = w0 * f0[c] + w1 * f1[c] + w2 * f2[c];
#pragma unroll 8
    for (int c = 0; c < 128; ++c)
        xrow[256 + c] = qrow[3 + c];

    float* xo = out_xyz + ((size_t)b * 4096 + q) * 3;
    xo[0] = qx; xo[1] = qy; xo[2] = qz;
}

// ---------------------------------------------------------------------------
// Kernel 2/3: D = relu(A(MxK,f32) * W(Kx256,f32) + bias), via
// V_WMMA_F32_16X16X4_F32 (native f32 matrix path, matches reference precision)
// block = 256 threads = 8 waves; block tile M=128, N=64; K staged in 32-wide
// LDS tiles (24 KB). Each wave: 16 rows x 64 cols = 4 v8f accumulators.
// grid = (M/128, 256/64). K must be a multiple of 32 (384 and 256 both are).
//
// f32 WMMA fragment layout (ISA 7.12.2):
//  A 16x4:  lanes 0-15 -> M=lane, {v0,v1}={K0,K1}; lanes 16-31 -> {K2,K3}
//  B 4x16:  v0: K=0 (lanes 0-15) / K=2 (lanes 16-31); v1: K=1 / K=3
//  C 16x16: VGPR r -> M=r (lanes 0-15) / M=r+8 (lanes 16-31), N=lane&15
//
// Bs is stored PAIR-INTERLEAVED so each lane's B fragment (two consecutive K
// values at one column) is one aligned 8-byte LDS load:
//   Bs[(k>>1)*128 + n*2 + (k&1)]  ->  fragment = v2f at (k>>1)*128 + n*2
// ---------------------------------------------------------------------------
__global__ void __launch_bounds__(256) gemm_bias_relu_wmma(
    const float* __restrict__ A,    // M x K, row-major
    const float* __restrict__ W,    // K x 256, row-major
    const float* __restrict__ bias, // 256
    float* __restrict__ D,          // M x 256, row-major
    int K)
{
    __shared__ float As[128 * 32];
    __shared__ float Bs[32 * 64];   // pair-interleaved, see above

    const int tid   = threadIdx.x;
    const int wave  = tid >> 5;
    const int lane  = tid & 31;
    const int laneM = lane & 15;   // row (A/C) or col (B/C) within 16
    const int hi    = lane >> 4;   // half-wave select

    const int blockM = blockIdx.x * 128;
    const int blockN = blockIdx.y * 64;

    v8f acc[4] = {};

    const int ktiles = K >> 5;
    for (int kt = 0; kt < ktiles; ++kt) {
        // stage A tile: 128x32 f32 = 1024 float4, 4 per thread (row-major)
#pragma unroll
        for (int i = 0; i < 4; ++i) {
            int lin = tid + i * 256;
            int row = lin >> 3, c4 = lin & 7;
            v4f v = *(const v4f*)(A + (size_t)(blockM + row) * K + kt * 32 + c4 * 4);
            *(v4f*)(As + row * 32 + c4 * 4) = v;
        }
        // stage B tile: 32x64 f32 = 512 float4, 2 per thread, pair-interleaved
#pragma unroll
        for (int i = 0; i < 2; ++i) {
            int lin = tid + i * 256;
            int row = lin >> 4, c4 = lin & 15;        // row = k within tile
            v4f v = *(const v4f*)(W + (size_t)(kt * 32 + row) * 256 + blockN + c4 * 4);
            float* dst = Bs + (row >> 1) * 128 + (row & 1);
            const int n0 = c4 * 4;
            dst[(n0 + 0) * 2] = v.x;
            dst[(n0 + 1) * 2] = v.y;
            dst[(n0 + 2) * 2] = v.z;
            dst[(n0 + 3) * 2] = v.w;
        }
        __syncthreads();

        const float* arow = As + (wave * 16 + laneM) * 32;
#pragma unroll
        for (int kk = 0; kk < 8; ++kk) {
            const int k = kk * 4 + 2 * hi;            // this half-wave's K pair
            v2f a = *(const v2f*)(arow + k);
            const float* bbase = Bs + (k >> 1) * 128 + laneM * 2;
#pragma unroll
            for (int j = 0; j < 4; ++j) {
                v2f bf = *(const v2f*)(bbase + j * 32);
                acc[j] = __builtin_amdgcn_wmma_f32_16x16x4_f32(
                    false, a, false, bf, (short)0, acc[j], false, false);
            }
        }
        __syncthreads();
    }

    // epilogue: bias + relu, scatter per C-layout
#pragma unroll
    for (int j = 0; j < 4; ++j) {
        const int col = blockN + j * 16 + laneM;
        const float bv = bias[col];
#pragma unroll
        for (int r = 0; r < 8; ++r) {
            const int row = blockM + wave * 16 + r + 8 * hi;
            D[(size_t)row * 256 + col] = fmaxf(acc[j][r] + bv, 0.0f);
        }
    }
}

// ---------------------------------------------------------------------------
extern "C" void kernel_launch(void* const* d_in, const int* in_sizes, int n_in,
                              void* d_out, int out_size, void* d_ws, size_t ws_size,
                              hipStream_t stream) {
    const float* pc1 = (const float*)d_in[0];  // 8*1024*259
    const float* pc2 = (const float*)d_in[1];  // 8*4096*131
    const float* W1  = (const float*)d_in[2];  // 384*256
    const float* b1  = (const float*)d_in[3];  // 256
    const float* W2  = (const float*)d_in[4];  // 256*256
    const float* b2  = (const float*)d_in[5];  // 256

    const size_t M = (size_t)8 * 4096;         // 32768 rows

    float* out_h   = (float*)d_out;            // 32768*256
    float* out_xyz = out_h + M * 256;          // 32768*3

    float* X = (float*)d_ws;                   // 32768 x 384 (50.3 MB)
    float* H = X + M * 384;                    // 32768 x 256 (33.5 MB)

    dim3 g1(4096 / 256, 8);
    knn_interp_kernel<<<g1, 256, 0, stream>>>(pc1, pc2, X, out_xyz);

    dim3 g2((unsigned)(M / 128), 256 / 64);    // (256, 4)
    gemm_bias_relu_wmma<<<g2, 256, 0, stream>>>(X, W1, b1, H, 384);
    gemm_bias_relu_wmma<<<g2, 256, 0, stream>>>(H, W2, b2, out_h, 256);
}